// KGEModel_87316685128476
// MI455X (gfx1250) — compile-verified
//
#include <hip/hip_runtime.h>
#include <math.h>

#define NE 100000
#define NR 200
#define STT 512
#define ABS_ 256
#define RDIM 384
#define BATCH 8192
#define XCOLS 406          // 6 + 2*NR
#define GAMMA_ 12.0f
#define PI_ 3.14159265358979323846f

typedef float v2f __attribute__((ext_vector_type(2)));
typedef float v8f __attribute__((ext_vector_type(8)));

// ---------------------------------------------------------------------------
// Kernel W: materialize WT[n, k] = W'[k, n]  (256 x 512), where
//   W'[k,j], j<128 :  w_e[k, j]          (k<256)   | -w_e[k, j]          (k>=256)
//   W'[k,j], j>=128: -w_e[k+256, j-128]  (k<256)   |  w_e[k-256, j-128]  (k>=256)
// Transposed layout makes each WMMA B-fragment a contiguous 8B global_load_b64.
// ---------------------------------------------------------------------------
__global__ __launch_bounds__(256)
void kge_wprep(const float* __restrict__ w_e, float* __restrict__ WT)
{
    const int idx = blockIdx.x * 256 + threadIdx.x;   // 0 .. 256*512-1
    const int n   = idx >> 9;                          // output column 0..255
    const int k   = idx & 511;                         // K index 0..511
    const bool jlow = (n < 128);
    const int  col  = jlow ? n : (n - 128);
    const int  row  = jlow ? k : ((k + 256) & 511);
    const float sg  = (jlow == (k < 256)) ? 1.f : -1.f;
    WT[(size_t)n * 512 + k] = sg * w_e[row * 128 + col];
}

// ---------------------------------------------------------------------------
// Kernel A: gathered GEMM  P[b, 0:256] = e_emb[idx[b], 0:512] @ W'(512x256)
// Two 16x16 D-tiles per wave, each a K-loop of 128 x V_WMMA_F32_16X16X4_F32.
// A-fragments from LDS, B-fragments as contiguous b64 loads from WT.
// ---------------------------------------------------------------------------
__global__ __launch_bounds__(256)
void kge_gemm_wmma(const int* __restrict__ x,
                   const float* __restrict__ e_emb,
                   const float* __restrict__ WT,
                   float* __restrict__ SP, float* __restrict__ OP)
{
    __shared__ float As[16][516];   // 16 x 512 (+4 pad), ~33 KB of 320 KB LDS
    __shared__ int   eids[16];

    const int b0   = blockIdx.x * 16;
    const int side = blockIdx.y;                 // 0 = subject, 1 = object
    const int tid  = threadIdx.x;
    const int lane = tid & 31;
    const int wid  = tid >> 5;
    float* OUT = side ? OP : SP;

    if (tid < 16) eids[tid] = x[(size_t)(b0 + tid) * XCOLS + (side ? 2 : 0)];
    __syncthreads();

    // cooperative gather of the 16x512 A tile
    for (int it = 0; it < 32; ++it) {
        int flat = tid + (it << 8);
        int row  = flat >> 9;
        int col  = flat & 511;
        As[row][col] = e_emb[(size_t)eids[row] * STT + col];
    }
    __syncthreads();

    const int nloc = lane & 15;
    const int h    = lane >> 4;

    for (int ntile = wid; ntile < 16; ntile += 8) {
        const int j = ntile * 16 + nloc;             // output column 0..255
        const float* wtrow = WT + (size_t)j * 512;   // contiguous K for column j
        v8f c = {0.f, 0.f, 0.f, 0.f, 0.f, 0.f, 0.f, 0.f};

        for (int kk = 0; kk < 128; ++kk) {
            const int k0 = kk * 4 + 2 * h;           // K index of .x; .y is k0+1
            v2f a;
            a.x = As[nloc][k0];
            a.y = As[nloc][k0 + 1];
            const v2f bv = *(const v2f*)(wtrow + k0); // one global_load_b64

            c = __builtin_amdgcn_wmma_f32_16x16x4_f32(
                    false, a, false, bv, (short)0, c, false, false);
        }

        #pragma unroll
        for (int v = 0; v < 8; ++v) {                // D: M = v + 8*h, N = nloc
            int m = v + 8 * h;
            OUT[(size_t)(b0 + m) * 256 + j] = c[v];
        }
    }
}

// ---------------------------------------------------------------------------
// Kernel B: relational phase sums (the transcendental hot loop)
//   R[b, k]     = sum_n w_rp[r, n] * cos(c[b,n] * f[k])   k = 0..127
//   R[b, 128+k] = sum_n w_rp[r, n] * sin(c[b,n] * f[k])
// One wave per (b, side); 8 waves/block -> 4 batch rows per block.
// Context ints + w_rp rows staged in LDS so the inner loop is pure TRANS+FMA.
// ---------------------------------------------------------------------------
__global__ __launch_bounds__(256)
void kge_relphase(const int* __restrict__ x,
                  const float* __restrict__ w_rp,
                  float* __restrict__ SR, float* __restrict__ OR_)
{
    __shared__ float cbuf[4][2][NR];   // context values (as float), 4 b x 2 sides
    __shared__ float wbuf[4][NR];      // w_rp row per local b
    __shared__ int   rbuf[4];

    const int tid  = threadIdx.x;
    const int lane = tid & 31;
    const int wid  = tid >> 5;
    const int bb0  = blockIdx.x * 4;             // first batch row of this block

    if (tid < 4) rbuf[tid] = x[(size_t)(bb0 + tid) * XCOLS + 1];
    __syncthreads();

    // stage 4 x 400 context ints (as float) and 4 x 200 w_rp floats
    for (int flat = tid; flat < 4 * 2 * NR; flat += 256) {
        int bb   = flat / (2 * NR);
        int rest = flat - bb * (2 * NR);         // 0..399
        cbuf[bb][rest >= NR][rest % NR] =
            (float)x[(size_t)(bb0 + bb) * XCOLS + 6 + rest];
    }
    for (int flat = tid; flat < 4 * NR; flat += 256) {
        int bb = flat / NR;
        int n  = flat - bb * NR;
        wbuf[bb][n] = w_rp[(size_t)rbuf[bb] * NR + n];
    }
    __syncthreads();

    const int bb   = wid >> 1;                   // local batch row 0..3
    const int side = wid & 1;
    const int b    = bb0 + bb;

    const float L = 13.28771237954945f / 128.0f; // log2(10000)/128
    const float f0 = exp2f(-(float)(lane     ) * L);
    const float f1 = exp2f(-(float)(lane + 32) * L);
    const float f2 = exp2f(-(float)(lane + 64) * L);
    const float f3 = exp2f(-(float)(lane + 96) * L);

    float ac0 = 0.f, ac1 = 0.f, ac2 = 0.f, ac3 = 0.f;
    float as0 = 0.f, as1 = 0.f, as2 = 0.f, as3 = 0.f;

    for (int n = 0; n < NR; ++n) {
        const float cv = cbuf[bb][side][n];
        const float wv = wbuf[bb][n];
        const float a0 = cv * f0, a1 = cv * f1, a2 = cv * f2, a3 = cv * f3;
        ac0 += wv * __cosf(a0);  as0 += wv * __sinf(a0);
        ac1 += wv * __cosf(a1);  as1 += wv * __sinf(a1);
        ac2 += wv * __cosf(a2);  as2 += wv * __sinf(a2);
        ac3 += wv * __cosf(a3);  as3 += wv * __sinf(a3);
    }

    float* out = (side ? OR_ : SR) + (size_t)b * 256;
    out[lane      ] = ac0;  out[lane + 32] = ac1;
    out[lane + 64 ] = ac2;  out[lane + 96] = ac3;
    out[128 + lane      ] = as0;  out[128 + lane + 32] = as1;
    out[128 + lane + 64 ] = as2;  out[128 + lane + 96] = as3;
}

// ---------------------------------------------------------------------------
// Kernel C: final score. One wave per batch element.
// ---------------------------------------------------------------------------
__global__ __launch_bounds__(256)
void kge_score(const int* __restrict__ x,
               const float* __restrict__ e_emb, const float* __restrict__ r_emb,
               const float* __restrict__ d_frq, const float* __restrict__ d_phi,
               const float* __restrict__ d_amp, const float* __restrict__ m_frq,
               const float* __restrict__ m_phi, const float* __restrict__ m_amp,
               const float* __restrict__ SP, const float* __restrict__ OP,
               const float* __restrict__ SR, const float* __restrict__ OR_,
               float* __restrict__ out)
{
    const int lane = threadIdx.x & 31;
    const int wid  = threadIdx.x >> 5;
    const int b    = blockIdx.x * 8 + wid;

    const int* xr = x + (size_t)b * XCOLS;
    const int s = xr[0], r = xr[1], o = xr[2];
    const float dab = (float)xr[3];
    const float mab = (float)xr[4];
    const float scale = PI_ / sqrtf(6.0f / (float)(NR + RDIM)); // PI / EMB_RNG_R

    const float* rrow = r_emb + (size_t)r * RDIM;
    const float* se   = e_emb + (size_t)s * STT;
    const float* oe   = e_emb + (size_t)o * STT;

    float sum = 0.f;

    // |rot(s) - o| over 384 components (256 static + 128 temporal)
    #pragma unroll
    for (int ii = 0; ii < 12; ++ii) {
        const int i = lane + 32 * ii;
        const float pr = rrow[i] * scale;
        const float rr = __cosf(pr), ri = __sinf(pr);
        float res, ims, reo, imo;
        if (ii < 8) {                       // i < 256: static entity embedding
            res = se[i];        ims = se[256 + i];
            reo = oe[i];        imo = oe[256 + i];
        } else {                            // temporal embedding component
            const int j = i - 256;          // 0..127
            const float sdA = d_amp[(size_t)s * 128 + j];
            const float smA = m_amp[(size_t)s * 128 + j];
            res = sdA * __cosf(dab * d_frq[(size_t)s * 256 + j]       + d_phi[(size_t)s * 256 + j])
                + smA * __cosf(mab * m_frq[(size_t)s * 256 + j]       + m_phi[(size_t)s * 256 + j]);
            ims = sdA * __sinf(dab * d_frq[(size_t)s * 256 + 128 + j] + d_phi[(size_t)s * 256 + 128 + j])
                + smA * __sinf(mab * m_frq[(size_t)s * 256 + 128 + j] + m_phi[(size_t)s * 256 + 128 + j]);
            const float odA = d_amp[(size_t)o * 128 + j];
            const float omA = m_amp[(size_t)o * 128 + j];
            reo = odA * __cosf(dab * d_frq[(size_t)o * 256 + j]       + d_phi[(size_t)o * 256 + j])
                + omA * __cosf(mab * m_frq[(size_t)o * 256 + j]       + m_phi[(size_t)o * 256 + j]);
            imo = odA * __sinf(dab * d_frq[(size_t)o * 256 + 128 + j] + d_phi[(size_t)o * 256 + 128 + j])
                + omA * __sinf(mab * m_frq[(size_t)o * 256 + 128 + j] + m_phi[(size_t)o * 256 + 128 + j]);
        }
        const float rsc = res * rr - ims * ri - reo;
        const float isc = res * ri + ims * rr - imo;
        sum += sqrtf(rsc * rsc + isc * isc);
    }

    // relational part: |(SP+SR) - (OP+OR)| over 128 complex components
    const float* sp = SP + (size_t)b * 256;  const float* sr  = SR  + (size_t)b * 256;
    const float* op = OP + (size_t)b * 256;  const float* orr = OR_ + (size_t)b * 256;
    #pragma unroll
    for (int jj = 0; jj < 4; ++jj) {
        const int j = lane + 32 * jj;
        const float sre = sp[j]       + sr[j];
        const float sim = sp[128 + j] + sr[128 + j];
        const float ore = op[j]       + orr[j];
        const float oim = op[128 + j] + orr[128 + j];
        const float dre = sre - ore, di = sim - oim;
        sum += sqrtf(dre * dre + di * di);
    }

    // wave32 reduction
    for (int off = 16; off > 0; off >>= 1)
        sum += __shfl_down(sum, off, 32);
    if (lane == 0) out[b] = GAMMA_ - sum;
}

// ---------------------------------------------------------------------------
extern "C" void kernel_launch(void* const* d_in, const int* in_sizes, int n_in,
                              void* d_out, int out_size, void* d_ws, size_t ws_size,
                              hipStream_t stream)
{
    const int*   x     = (const int*)  d_in[0];
    const float* e_emb = (const float*)d_in[1];
    const float* r_emb = (const float*)d_in[2];
    const float* d_frq = (const float*)d_in[3];
    const float* d_phi = (const float*)d_in[4];
    const float* d_amp = (const float*)d_in[5];
    const float* m_frq = (const float*)d_in[6];
    const float* m_phi = (const float*)d_in[7];
    const float* m_amp = (const float*)d_in[8];
    const float* w_e   = (const float*)d_in[9];
    const float* w_rp  = (const float*)d_in[10];
    float* out = (float*)d_out;

    // workspace: SP | OP | SR | OR (8 MB each) | WT (512 KB)
    float* SP  = (float*)d_ws;
    float* OP  = SP + (size_t)BATCH * 256;
    float* SR  = OP + (size_t)BATCH * 256;
    float* OR_ = SR + (size_t)BATCH * 256;
    float* WT  = OR_ + (size_t)BATCH * 256;

    hipLaunchKernelGGL(kge_wprep, dim3(256 * 512 / 256), dim3(256), 0, stream,
                       w_e, WT);
    dim3 gA(BATCH / 16, 2);
    hipLaunchKernelGGL(kge_gemm_wmma, gA, dim3(256), 0, stream,
                       x, e_emb, WT, SP, OP);
    hipLaunchKernelGGL(kge_relphase, dim3(BATCH / 4), dim3(256), 0, stream,
                       x, w_rp, SR, OR_);
    hipLaunchKernelGGL(kge_score, dim3(BATCH / 8), dim3(256), 0, stream,
                       x, e_emb, r_emb, d_frq, d_phi, d_amp,
                       m_frq, m_phi, m_amp, SP, OP, SR, OR_, out);
}